// FlowReconstructionLossV1_4612794875985
// MI455X (gfx1250) — compile-verified
//
#include <hip/hip_runtime.h>

// FlowReconstructionLossV1 for MI455X (gfx1250, wave32).
// Memory-bound warp+L1 loss: coalesced streaming of flow/im2, cache-local
// bilinear gathers from im1, and WMMA-based (v_wmma_f32_16x16x4_f32)
// deterministic reduction trees.

typedef float v2f __attribute__((ext_vector_type(2)));
typedef float v8f __attribute__((ext_vector_type(8)));

#define IMG_B 16
#define IMG_C 3
#define IMG_H 512
#define IMG_W 1024
#define HW    (IMG_H * IMG_W)          // 524288 = 2^19
#define NPIX  (IMG_B * HW)             // 8388608
#define NBLK  4096
#define TPB   256                      // 8 waves of 32

// Full-wave (32 lane) sum via one f32 WMMA against a ones matrix.
// A is 16x4 f32: per-lane {s, 0} -> A[m][0]=s_m (lanes 0-15), A[m][2]=s_{m+16}.
// D[m][n] = sum_k A[m][k] (B==1), identical across columns. C/D layout:
// lanes 0-15 hold rows 0-7, lanes 16-31 hold rows 8-15, so summing the 8
// accumulator VGPRs and xor-shuffling across the half-wave yields the full
// 32-lane sum in every lane. Pure f32 accumulation, fixed order -> deterministic.
__device__ __forceinline__ float wave_reduce32(float s) {
    v2f a; a[0] = s;     a[1] = 0.0f;
    v2f b; b[0] = 1.0f;  b[1] = 1.0f;
    v8f c = {};
    c = __builtin_amdgcn_wmma_f32_16x16x4_f32(
        /*neg_a=*/false, a, /*neg_b=*/false, b,
        /*c_mod=*/(short)0, c, /*reuse_a=*/false, /*reuse_b=*/false);
    float h = ((c[0] + c[1]) + (c[2] + c[3])) + ((c[4] + c[5]) + (c[6] + c[7]));
    h += __shfl_xor(h, 16, 32);
    return h;
}

// Block-wide sum. Every wave executes the WMMA with full EXEC (no divergent
// guard around it); result is valid in all lanes of all waves.
__device__ __forceinline__ float block_reduce(float s, float* lds) {
    float w = wave_reduce32(s);
    const int lane = threadIdx.x & 31;
    const int wid  = threadIdx.x >> 5;
    if (lane == 0) lds[wid] = w;
    __syncthreads();
    float v = (lane < (TPB / 32)) ? lds[lane] : 0.0f;
    return wave_reduce32(v);
}

__global__ void __launch_bounds__(TPB)
flow_loss_partial(const float2* __restrict__ flow,
                  const float*  __restrict__ im1,
                  const float*  __restrict__ im2,
                  float*        __restrict__ partial) {
    __shared__ float lds[TPB / 32];
    float acc = 0.0f;

    const int stride = gridDim.x * TPB;
    for (int i = blockIdx.x * TPB + threadIdx.x; i < NPIX; i += stride) {
        // i -> (b, y, x); HW = 2^19, W = 2^10
        const int b = i >> 19;
        const int y = (i >> 10) & (IMG_H - 1);
        const int x = i & (IMG_W - 1);

        const float2 f = flow[i];                 // coalesced 8B load
        // ix = ((gx+1)*W - 1)/2 with gx = 2x/W - 1 + fx/W  ==>  x - 0.5 + fx/2
        const float ix = (float)x - 0.5f + 0.5f * f.x;
        const float iy = (float)y - 0.5f + 0.5f * f.y;

        const float x0f = floorf(ix), y0f = floorf(iy);
        const float wx1 = ix - x0f,   wy1 = iy - y0f;
        const float wx0 = 1.0f - wx1, wy0 = 1.0f - wy1;

        const int x0 = (int)x0f, y0 = (int)y0f;
        const int x1 = x0 + 1,   y1 = y0 + 1;

        const bool vx0 = (unsigned)x0 < IMG_W, vx1 = (unsigned)x1 < IMG_W;
        const bool vy0 = (unsigned)y0 < IMG_H, vy1 = (unsigned)y1 < IMG_H;

        const float w00 = wx0 * wy0 * (float)(vx0 && vy0);
        const float w10 = wx1 * wy0 * (float)(vx1 && vy0);
        const float w01 = wx0 * wy1 * (float)(vx0 && vy1);
        const float w11 = wx1 * wy1 * (float)(vx1 && vy1);

        const int x0c = max(0, min(x0, IMG_W - 1));
        const int x1c = max(0, min(x1, IMG_W - 1));
        const int y0c = max(0, min(y0, IMG_H - 1));
        const int y1c = max(0, min(y1, IMG_H - 1));

        const int base = b * (IMG_C * HW);
        const int o00 = y0c * IMG_W + x0c;
        const int o10 = y0c * IMG_W + x1c;
        const int o01 = y1c * IMG_W + x0c;
        const int o11 = y1c * IMG_W + x1c;
        const int oc  = y * IMG_W + x;

        #pragma unroll
        for (int ch = 0; ch < IMG_C; ++ch) {
            const float* p1 = im1 + base + ch * HW;
            const float v = w00 * p1[o00] + w10 * p1[o10]
                          + w01 * p1[o01] + w11 * p1[o11];
            acc += fabsf(im2[base + ch * HW + oc] - v);
        }
    }

    const float bs = block_reduce(acc, lds);
    if (threadIdx.x == 0) partial[blockIdx.x] = bs;
}

__global__ void __launch_bounds__(TPB)
flow_loss_final(const float* __restrict__ partial, float* __restrict__ out) {
    __shared__ float lds[TPB / 32];
    float s = 0.0f;
    #pragma unroll
    for (int i = threadIdx.x; i < NBLK; i += TPB) s += partial[i];
    const float t = block_reduce(s, lds);
    if (threadIdx.x == 0) out[0] = t * (1.0f / (float)HW);
}

extern "C" void kernel_launch(void* const* d_in, const int* in_sizes, int n_in,
                              void* d_out, int out_size, void* d_ws, size_t ws_size,
                              hipStream_t stream) {
    const float2* flow = (const float2*)d_in[0];   // (B,H,W,2) f32
    const float*  im1  = (const float*)d_in[1];    // (B,C,H,W) f32
    const float*  im2  = (const float*)d_in[2];    // (B,C,H,W) f32
    float* partial = (float*)d_ws;                 // NBLK floats = 16 KB
    float* out     = (float*)d_out;                // scalar f32

    flow_loss_partial<<<NBLK, TPB, 0, stream>>>(flow, im1, im2, partial);
    flow_loss_final<<<1, TPB, 0, stream>>>(partial, out);
}